// CustomLossTorch_83820581748769
// MI455X (gfx1250) — compile-verified
//
#include <hip/hip_runtime.h>
#include <math.h>

// ---------------------------------------------------------------------------
// Block-local complex projection loss for MI455X (gfx1250, wave32, WMMA).
// 2048 workgroups, one per (b, block_t, block_x). f32 WMMA 16x16x4 for all
// Gram / rhs / reconstruction matmuls; LDS-resident panels (16 rows, rows
// 12..15 zero -> branch/mask-free WMMA fragments); G_im built as T - T^T at
// solve time (saves one WMMA chain); wave-synchronous complex Gauss-Jordan;
// global f32 atomics for the cross-block norm.
// ---------------------------------------------------------------------------

typedef __attribute__((ext_vector_type(2))) float v2f;
typedef __attribute__((ext_vector_type(8))) float v8f;

#define NVEC   12      // NV == NVt
#define NPAD   16      // padded row count (rows 12..15 zeroed)
#define DDEPTH 512     // S*bt*bx = 2*16*16
#define LSTR   516     // LDS row stride (512 + 4): float4-aligned, bank-spread
#define TSTR   17      // 16x16 reduce-tile row stride (kills half-vs-half conflicts)

// float offsets into dynamic LDS
#define OFF_PR   0
#define OFF_PI   (OFF_PR + NPAD*LSTR)      //  8256
#define OFF_VR   (OFF_PI + NPAD*LSTR)      // 16512
#define OFF_VI   (OFF_VR + NPAD*LSTR)      // 24768
#define OFF_TGRE (OFF_VI + NPAD*LSTR)      // 33024
#define OFF_TGIM (OFF_TGRE + 16*TSTR)      // holds T = Pr^T Pi (G_im = T - T^T)
#define OFF_TRRE (OFF_TGIM + 16*TSTR)
#define OFF_TRIM (OFF_TRRE + 16*TSTR)
#define OFF_CR   (OFF_TRIM + 16*TSTR)      // coeff re, 12x16 (cols 12..15 = 0)
#define OFF_CI   (OFF_CR + 12*16)
#define OFF_ACCV (OFF_CI + 12*16)          // 12 per-v squared-residual sums
#define ZERO_BEG OFF_TGRE
#define ZERO_CNT (OFF_ACCV + 12 - OFF_TGRE)
#define OFF_AGR  (OFF_ACCV + 12)           // augmented [G|rhs] re, 12 x 28
#define OFF_AGI  (OFF_AGR + 12*28)
#define LDS_FLOATS (OFF_AGI + 12*28)       // 35180 floats = 140720 bytes

__device__ __forceinline__ v8f wmma_f32(v2f a, v2f b, v8f c) {
  // D = A(16x4) * B(4x16) + C, all f32.  8-arg form:
  // (neg_a, A, neg_b, B, c_mod, C, reuse_a, reuse_b)
  return __builtin_amdgcn_wmma_f32_16x16x4_f32(false, a, false, b,
                                               (short)0, c, false, false);
}

// Fragment for A^T-style AND plain-B-style operand of a D-chunk matmul:
// lane l (m = l&15, half = l>>4) holds M[m][db + 2*half + {0,1}].
// Rows 12..15 are zeroed in LDS -> no masking, EXEC stays uniform.
__device__ __forceinline__ v2f ldfragT(const float* M, int db, int lane) {
  const int a = (lane & 15) * LSTR + db + 2 * (lane >> 4);
  v2f f;
  f.x = M[a];
  f.y = M[a + 1];
  return f;
}

// Cooperative load of one (b, block) panel: global (n,s,t,x) -> LDS [n][d],
// d = s*256 + t*16 + x.  float4 loads (16B aligned), float4 LDS stores.
// Also zeroes pad rows 12..15 so WMMA fragments need no masking.
__device__ __forceinline__ void load_panel(float* dst, const float* src,
                                           int b, int it, int jx, int tid) {
  for (int i = tid * 4; i < (NPAD - NVEC) * LSTR; i += 256 * 4)
    *reinterpret_cast<float4*>(dst + NVEC * LSTR + i) = float4{0.f, 0.f, 0.f, 0.f};
  for (int e = tid; e < NVEC * 2 * 16 * 4; e += 256) {  // 1536 float4
    int x4  = e & 3;
    int row = e >> 2;            // 0..383
    int t   = row & 15;
    int s   = (row >> 4) & 1;
    int n   = row >> 5;
    int gidx = ((((b * NVEC + n) * 2 + s) * 256) + (it * 16 + t)) * 256
               + jx * 16 + x4 * 4;
    float4 val = *reinterpret_cast<const float4*>(src + gidx);
    int d = s * 256 + t * 16 + x4 * 4;
    *reinterpret_cast<float4*>(dst + n * LSTR + d) = val;
  }
}

__global__ void __launch_bounds__(256)
proj_loss_kernel(const float* __restrict__ pred_re,
                 const float* __restrict__ pred_im,
                 const float* __restrict__ targ_re,
                 const float* __restrict__ targ_im,
                 float* __restrict__ ws) {
  extern __shared__ float sm[];
  const int tid  = threadIdx.x;
  const int lane = tid & 31;
  const int wave = tid >> 5;

  const int bid = blockIdx.x;      // 0..2047
  const int b   = bid >> 8;        // / 256
  const int rem = bid & 255;
  const int it  = rem >> 4;        // block_t
  const int jx  = rem & 15;        // block_x

  // ---------------- Stage 1: load panels + zero reduce scratch -------------
  for (int i = tid; i < ZERO_CNT; i += 256) sm[ZERO_BEG + i] = 0.0f;
  load_panel(sm + OFF_PR, pred_re, b, it, jx, tid);
  load_panel(sm + OFF_PI, pred_im, b, it, jx, tid);
  load_panel(sm + OFF_VR, targ_re, b, it, jx, tid);
  load_panel(sm + OFF_VI, targ_im, b, it, jx, tid);
  __syncthreads();

  // ---------------- Stage 2: G = P^H P, rhs = P^H V via f32 WMMA -----------
  // Each wave owns 16 of the 128 K-chunks (K=4 each), accumulates 16x16
  // f32 tiles, then reduces through LDS ds_add_f32.  G_im needs only
  // T = Pr^T Pi here (G_im = T - T^T, applied at the solve stage).
  {
    v8f gRe = {}, gImT = {}, rRe = {}, rImA = {}, rImB = {};
    const int kc0 = wave * 16;
    for (int kc = kc0; kc < kc0 + 16; ++kc) {
      const int db = kc * 4;
      v2f fPr = ldfragT(sm + OFF_PR, db, lane);
      v2f fPi = ldfragT(sm + OFF_PI, db, lane);
      v2f fVr = ldfragT(sm + OFF_VR, db, lane);
      v2f fVi = ldfragT(sm + OFF_VI, db, lane);
      gRe  = wmma_f32(fPr, fPr, gRe);    // Pr^T Pr
      gRe  = wmma_f32(fPi, fPi, gRe);    // + Pi^T Pi
      gImT = wmma_f32(fPr, fPi, gImT);   // T = Pr^T Pi
      rRe  = wmma_f32(fPr, fVr, rRe);    // Pr^T Vr
      rRe  = wmma_f32(fPi, fVi, rRe);    // + Pi^T Vi
      rImA = wmma_f32(fPr, fVi, rImA);   // Pr^T Vi
      rImB = wmma_f32(fPi, fVr, rImB);   // Pi^T Vr   (rhs_im = A - B)
    }
    const int N = lane & 15, half = lane >> 4;
#pragma unroll
    for (int r = 0; r < 8; ++r) {
      const int Mr = r + 8 * half;       // C layout: VGPR r, halves = M, M+8
      const int o  = Mr * TSTR + N;
      __hip_atomic_fetch_add(&sm[OFF_TGRE + o], gRe[r],
                             __ATOMIC_RELAXED, __HIP_MEMORY_SCOPE_WORKGROUP);
      __hip_atomic_fetch_add(&sm[OFF_TGIM + o], gImT[r],
                             __ATOMIC_RELAXED, __HIP_MEMORY_SCOPE_WORKGROUP);
      __hip_atomic_fetch_add(&sm[OFF_TRRE + o], rRe[r],
                             __ATOMIC_RELAXED, __HIP_MEMORY_SCOPE_WORKGROUP);
      __hip_atomic_fetch_add(&sm[OFF_TRIM + o], rImA[r] - rImB[r],
                             __ATOMIC_RELAXED, __HIP_MEMORY_SCOPE_WORKGROUP);
    }
  }
  __syncthreads();

  // ---------------- Stage 3: solve (G + eps I) coeff = rhs -----------------
  // 12x12 complex Hermitian PD, 12 RHS. Wave-0-only Gauss-Jordan; LDS ops
  // from one wave are processed in order, so no intra-wave barriers needed.
  if (wave == 0) {
    for (int e = lane; e < 12 * 24; e += 32) {
      int r = e / 24, c = e % 24;
      float re, im;
      if (c < 12) {
        re = sm[OFF_TGRE + r * TSTR + c];
        im = sm[OFF_TGIM + r * TSTR + c]     // G_im = T - T^T
           - sm[OFF_TGIM + c * TSTR + r];
        if (c == r) re += 1e-6f;             // EPS * I (real diagonal)
      } else {
        re = sm[OFF_TRRE + r * TSTR + (c - 12)];
        im = sm[OFF_TRIM + r * TSTR + (c - 12)];
      }
      sm[OFF_AGR + r * 28 + c] = re;
      sm[OFF_AGI + r * 28 + c] = im;
    }
    const int j = lane;                  // lanes 0..23 own augmented columns
    for (int k = 0; k < 12; ++k) {
      float pre = sm[OFF_AGR + k * 28 + k];
      float pim = sm[OFF_AGI + k * 28 + k];
      float inv = 1.0f / (pre * pre + pim * pim + 1e-30f);
      float ire =  pre * inv;
      float iim = -pim * inv;
      float rkr = 0.0f, rki = 0.0f;
      if (j < 24) {
        float ar = sm[OFF_AGR + k * 28 + j];
        float ai = sm[OFF_AGI + k * 28 + j];
        rkr = ar * ire - ai * iim;
        rki = ar * iim + ai * ire;
        sm[OFF_AGR + k * 28 + j] = rkr;
        sm[OFF_AGI + k * 28 + j] = rki;
      }
      for (int r = 0; r < 12; ++r) {
        if (r == k) continue;
        float fr = sm[OFF_AGR + r * 28 + k];   // read before row update
        float fi = sm[OFF_AGI + r * 28 + k];
        if (j < 24) {
          sm[OFF_AGR + r * 28 + j] -= fr * rkr - fi * rki;
          sm[OFF_AGI + r * 28 + j] -= fr * rki + fi * rkr;
        }
      }
    }
    for (int e = lane; e < 12 * 12; e += 32) {  // cols 12..15 stay zero
      int n = e / 12, v = e % 12;
      sm[OFF_CR + n * 16 + v] = sm[OFF_AGR + n * 28 + 12 + v];
      sm[OFF_CI + n * 16 + v] = sm[OFF_AGI + n * 28 + 12 + v];
    }
  }
  __syncthreads();

  // ------------- Stage 4+5: corrected = P coeff, residual norms ------------
  // 32 row-tiles of 16 d-rows; each wave owns 4. K = 12 -> 3 WMMA k-chunks.
  {
    float vsum = 0.0f;
    const int m = lane & 15, half = lane >> 4;
    for (int t = 0; t < 4; ++t) {
      const int d0 = (wave * 4 + t) * 16;
      v8f aR = {}, aI = {};
#pragma unroll
      for (int c = 0; c < 3; ++c) {
        const int r0 = c * 4 + 2 * half;   // coeff row for this lane-half
        v2f fPr, fPi, bCr, bCi;
        fPr.x = sm[OFF_PR + r0 * LSTR + d0 + m];
        fPr.y = sm[OFF_PR + (r0 + 1) * LSTR + d0 + m];
        fPi.x = sm[OFF_PI + r0 * LSTR + d0 + m];
        fPi.y = sm[OFF_PI + (r0 + 1) * LSTR + d0 + m];
        bCr.x = sm[OFF_CR + r0 * 16 + m];
        bCr.y = sm[OFF_CR + (r0 + 1) * 16 + m];
        bCi.x = sm[OFF_CI + r0 * 16 + m];
        bCi.y = sm[OFF_CI + (r0 + 1) * 16 + m];
        v2f bCiN = -bCi;                   // f32 WMMA has no A/B neg: do it here
        aR = wmma_f32(fPr, bCr, aR);       // Pr * cr
        aR = wmma_f32(fPi, bCiN, aR);      // - Pi * ci
        aI = wmma_f32(fPr, bCi, aI);       // Pr * ci
        aI = wmma_f32(fPi, bCr, aI);       // + Pi * cr
      }
      if (m < NVEC) {                      // lane holds (d0 + r + 8*half, v=m)
#pragma unroll
        for (int r = 0; r < 8; ++r) {
          const int d = d0 + r + 8 * half;
          float vr = sm[OFF_VR + m * LSTR + d];
          float vi = sm[OFF_VI + m * LSTR + d];
          float dr = vr - aR[r];
          float di = vi - aI[r];
          vsum += dr * dr + di * di;
        }
      }
    }
    if ((lane & 15) < NVEC)
      __hip_atomic_fetch_add(&sm[OFF_ACCV + (lane & 15)], vsum,
                             __ATOMIC_RELAXED, __HIP_MEMORY_SCOPE_WORKGROUP);
  }
  __syncthreads();

  if (tid < NVEC)
    __hip_atomic_fetch_add(&ws[b * NVEC + tid], sm[OFF_ACCV + tid],
                           __ATOMIC_RELAXED, __HIP_MEMORY_SCOPE_AGENT);
}

__global__ void zero_ws_kernel(float* ws) {
  if (threadIdx.x < 96) ws[threadIdx.x] = 0.0f;
}

__global__ void finalize_kernel(const float* __restrict__ ws,
                                float* __restrict__ out) {
  __shared__ float red[128];
  const int t = threadIdx.x;
  red[t] = (t < 96) ? sqrtf(ws[t]) : 0.0f;
  __syncthreads();
  for (int s = 64; s > 0; s >>= 1) {
    if (t < s) red[t] += red[t + s];
    __syncthreads();
  }
  if (t == 0) out[0] = red[0] / 96.0f;
}

extern "C" void kernel_launch(void* const* d_in, const int* in_sizes, int n_in,
                              void* d_out, int out_size, void* d_ws,
                              size_t ws_size, hipStream_t stream) {
  const float* pred_re = (const float*)d_in[0];
  const float* pred_im = (const float*)d_in[1];
  const float* targ_re = (const float*)d_in[2];
  const float* targ_im = (const float*)d_in[3];
  float* ws  = (float*)d_ws;
  float* out = (float*)d_out;

  zero_ws_kernel<<<1, 128, 0, stream>>>(ws);
  proj_loss_kernel<<<2048, 256, LDS_FLOATS * sizeof(float), stream>>>(
      pred_re, pred_im, targ_re, targ_im, ws);
  finalize_kernel<<<1, 128, 0, stream>>>(ws, out);
}